// HiPPO_59055800320637
// MI455X (gfx1250) — compile-verified
//
#include <hip/hip_runtime.h>

// y[m, n] = sum_l U[m, l] * K[n, T-1-l]   (M=512, T=1024, N=64)
// Grid = 32 M-tiles x 4 N-tiles; block = 256 threads = 8 K-split waves.
// Wave 0 DMAs the 16x1024 f32 U-tile into LDS via the Tensor Data Mover
// (tensor_load_to_lds, TENSORcnt) with TDM padding (1 DWORD per 256) for
// bank-conflict-free A reads. All threads cooperatively stage the
// time-REVERSED K-tile into LDS (pitch 1028), so B reads are forward,
// aligned ds_load_b64 with no component swap. Inner loop: DS loads +
// v_wmma_f32_16x16x4_f32 with a compile-time trip count (no exec masking).
// K-split partials reduce deterministically via LDS (aliasing the dead U tile).

typedef __attribute__((ext_vector_type(2))) float v2f;
typedef __attribute__((ext_vector_type(4))) float v4f;
typedef __attribute__((ext_vector_type(8))) float v8f;
typedef __attribute__((ext_vector_type(4))) unsigned int u32x4;
typedef __attribute__((ext_vector_type(8))) unsigned int u32x8;

#define T_LEN  1024
#define N_ORD  64
#define M_ROWS 512
#define PITCH  1028   // 1024 data DWORDs + 4 pad DWORDs per row (banks rotate by 4)

__global__ __launch_bounds__(256) void hippo_gemm_wmma_f32(
    const float* __restrict__ U,   // (512, 1024)  inputs flattened (b,e,d)
    const float* __restrict__ Km,  // (64, 1024)   HiPPO kernel
    float* __restrict__ Y)         // (512, 64)
{
    __shared__ float uTile[16 * PITCH]; // TDM-staged U tile (intra-row pads)
    __shared__ float kTile[16 * PITCH]; // software-staged reversed K (row pitch 1028)

    const int tid    = threadIdx.x;
    const int lane   = tid & 31;
    const int kSplit = tid >> 5;        // wave id 0..7 -> l in [128*kSplit, +128)
    const int m0     = blockIdx.x * 16; // gridDim.x = 32
    const int n0     = blockIdx.y * 16; // gridDim.y = 4

    // ---- TDM: DMA U[m0..m0+15, 0..1023] (f32) into LDS, padded pitch ----
    if (kSplit == 0) {
        const unsigned long long ga =
            (unsigned long long)(uintptr_t)(U + (size_t)m0 * T_LEN);
        const unsigned ldsAddr = (unsigned)(uintptr_t)(void*)uTile;

        u32x4 g0;                                    // D# group 0 (§8.3)
        g0.x = 1u;                                   // count=1 (valid), user mode
        g0.y = ldsAddr;                              // lds_addr (bytes)
        g0.z = (unsigned)(ga & 0xffffffffull);       // global_addr[31:0]
        g0.w = (unsigned)((ga >> 32) & 0x01ffffffu)  // global_addr[56:32]
               | (2u << 30);                         // type=2 ("image")

        u32x8 g1;                                    // D# group 1 (§8.4)
        g1.s0 = (2u << 16)        // data_size = 4 bytes
              | (1u << 20)        // pad_enable
              | (7u << 22)        // pad_interval: every 256 DWORDs
              | (0u << 25);       // pad_amount:   1 DWORD
        g1.s1 = (unsigned)((T_LEN & 0xffffu) << 16); // tensor_dim0[15:0]
        g1.s2 = (unsigned)(M_ROWS << 16);            // dim0 hi=0 | tensor_dim1 lo
        g1.s3 = (unsigned)(T_LEN << 16);             // dim1 hi=0 | tile_dim0=1024
        g1.s4 = 16u;                                 // tile_dim1=16, tile_dim2=0
        g1.s5 = (unsigned)T_LEN;                     // tensor_dim0_stride[31:0]
        g1.s6 = 0u;
        g1.s7 = 0u;

        asm volatile("tensor_load_to_lds %0, %1" :: "s"(g0), "s"(g1) : "memory");
    }

    // ---- Cooperative staging of reversed K tile (overlaps with the TDM) ----
    // Row `it` of K[n0..n0+15] loaded as coalesced float4 (256 thr * 16B = one
    // 4KB row per iteration), component-reversed, stored 16B-aligned into
    // kTile[row*PITCH + (1020 - c) ...] so B fragments read forward.
    {
        const float* kSrc = Km + (size_t)n0 * T_LEN;
        #pragma unroll
        for (int it = 0; it < 16; ++it) {
            const int c = tid * 4;                       // 0..1020
            v4f f = *(const v4f*)(kSrc + it * T_LEN + c);
            v4f r = { f.w, f.z, f.y, f.x };
            *(v4f*)(kTile + it * PITCH + (1020 - c)) = r;
        }
    }

    __builtin_amdgcn_s_wait_tensorcnt(0);  // meaningful in wave 0; no-op elsewhere
    __syncthreads();

    // ---- WMMA f32 16x16x4 operand layout (ISA 7.12.2) ----
    //   A (16x4): lanes 0-15: M=lane, v0/v1 = K 0/1; lanes 16-31: K 2/3.
    //   B (4x16): mirrored; N = lane % 16.
    const int half = lane >> 4;        // 0 or 1
    const int idx  = lane & 15;        // M row (A) / N col (B)
    const int kOff = half * 2;         // 0 or 2

    // TDM-padded element (row, l) = row*PITCH + l + (l>>8); within this wave's
    // 128-long l-range, (l>>8) == (kSplit>>1), a compile-time-foldable constant.
    // Fold the wave's lBeg into the base pointers so the compute loop has a
    // literal trip count (uniform scalar loop control, no exec masking).
    const int lBeg = kSplit * 128;
    const float* aPtr = uTile + idx * PITCH + (kSplit >> 1) + kOff + lBeg;
    const float* bPtr = kTile + idx * PITCH + kOff + lBeg;

    v8f c = {0.f, 0.f, 0.f, 0.f, 0.f, 0.f, 0.f, 0.f};

    #pragma unroll 8
    for (int j = 0; j < 128; j += 4) {
        v2f a = { aPtr[j], aPtr[j + 1] };       // ds_load_2addr_b32
        v2f b = *(const v2f*)(bPtr + j);        // ds_load_b64 (8B aligned)
        c = __builtin_amdgcn_wmma_f32_16x16x4_f32(
                false, a, false, b, (short)0, c, false, false);
    }

    // All reads of the U tile must finish before partials overwrite it.
    __syncthreads();

    // Deterministic K-split reduction (fixed order): partials from kSplit 1..7
    // stored into the first 7 KB of uTile, conflict-free (r*32 + lane).
    float* red = uTile;                               // [7][256] floats
    if (kSplit != 0) {
        float* dst = red + (kSplit - 1) * 256;
        #pragma unroll
        for (int r = 0; r < 8; ++r) dst[r * 32 + lane] = c[r];
    }
    __syncthreads();

    if (kSplit == 0) {
        #pragma unroll
        for (int s = 0; s < 7; ++s) {
            const float* src = red + s * 256;
            #pragma unroll
            for (int r = 0; r < 8; ++r) c[r] += src[r * 32 + lane];
        }
        // C/D layout: VGPR r holds M = m0 + r + 8*half, N = n0 + idx.
        #pragma unroll
        for (int r = 0; r < 8; ++r) {
            Y[(m0 + r + 8 * half) * N_ORD + (n0 + idx)] = c[r];
        }
    }
}

extern "C" void kernel_launch(void* const* d_in, const int* in_sizes, int n_in,
                              void* d_out, int out_size, void* d_ws, size_t ws_size,
                              hipStream_t stream) {
    (void)in_sizes; (void)n_in; (void)out_size; (void)d_ws; (void)ws_size;
    const float* U  = (const float*)d_in[0];   // (4,2,64,1024) f32 -> (512,1024)
    const float* Km = (const float*)d_in[1];   // (64,1024) f32
    float*       Y  = (float*)d_out;           // (4,2,64,64) f32 -> (512,64)

    hippo_gemm_wmma_f32<<<dim3(M_ROWS / 16, N_ORD / 16), dim3(256), 0, stream>>>(U, Km, Y);
}